// MassSpringSolver_39788577030968
// MI455X (gfx1250) — compile-verified
//
#include <hip/hip_runtime.h>
#include <math.h>

// Problem constants (match reference)
#define BB        4
#define NVV       100000
#define NEE       400000
#define NSUB      10
#define TRAJ      (NSUB + 1)
#define DT_C      0.01f
#define K_SPRING  1000.0f
#define DAMP_C    0.999f
#define ACT_SCALE 0.1f
#define EPS_C     1e-6f
#define GRAV_Y    (-9.8f)

#define SLICE     (NVV * 3)       // floats per (batch, substep) slice
#define SLICE4    (SLICE / 4)     // 75000, exact

// Native f32 atomic add (CDNA5 GLOBAL/FLAT_ATOMIC_ADD_F32), relaxed, device scope,
// result discarded -> non-returning atomic tracked with STOREcnt.
__device__ __forceinline__ void atom_add_f32(float* p, float v) {
    __hip_atomic_fetch_add(p, v, __ATOMIC_RELAXED, __HIP_MEMORY_SCOPE_AGENT);
}

// ---- Never launched. Defined FIRST purely so the disasm snippet shows the
// ---- exact lowering of the f32 atomic add (native vs CAS loop).
__global__ void atomic_lowering_probe(float* __restrict__ p, float v) {
    atom_add_f32(p + threadIdx.x, v);
}

// One thread per edge, blockIdx.y = batch: gather endpoints from trajectory
// slice s, compute spring force, scatter-add with native f32 atomics.
__global__ void force_kernel(const float* __restrict__ traj_pos,
                             const int2* __restrict__ edges,
                             const float* __restrict__ rest_eff,
                             float* __restrict__ f_node,
                             int s) {
    int e = blockIdx.x * blockDim.x + threadIdx.x;
    if (e >= NEE) return;
    int b = blockIdx.y;

    // Stream-ahead prefetch (OOB-safe, speculative): gfx1250 global_prefetch_b8,
    // locality 3 -> WGP scope (fills all cache levels on miss).
    __builtin_prefetch(&edges[e] + 4096, 0, 3);
    __builtin_prefetch(&rest_eff[(size_t)b * NEE + e] + 4096, 0, 3);

    int2 ij = edges[e];                                        // 8B load
    const float* pb = traj_pos + (size_t)(b * TRAJ + s) * SLICE;

    float pix = pb[3 * ij.x + 0];
    float piy = pb[3 * ij.x + 1];
    float piz = pb[3 * ij.x + 2];
    float pjx = pb[3 * ij.y + 0];
    float pjy = pb[3 * ij.y + 1];
    float pjz = pb[3 * ij.y + 2];

    float dx = pjx - pix, dy = pjy - piy, dz = pjz - piz;
    float len  = sqrtf(dx * dx + dy * dy + dz * dz + EPS_C);
    float coef = K_SPRING * (len - rest_eff[(size_t)b * NEE + e]) / len;
    float fx = coef * dx, fy = coef * dy, fz = coef * dz;

    float* fb = f_node + (size_t)b * SLICE;
    atom_add_f32(fb + 3 * ij.x + 0,  fx);
    atom_add_f32(fb + 3 * ij.x + 1,  fy);
    atom_add_f32(fb + 3 * ij.x + 2,  fz);
    atom_add_f32(fb + 3 * ij.y + 0, -fx);
    atom_add_f32(fb + 3 * ij.y + 1, -fy);
    atom_add_f32(fb + 3 * ij.y + 2, -fz);
}

// One thread per node, blockIdx.y = batch: semi-implicit Euler step on a full
// float3 (b96-mergeable loads/stores), writes slice s+1, re-zeroes f_node.
__global__ void integrate_kernel(float* __restrict__ traj_pos,
                                 float* __restrict__ traj_vel,
                                 float* __restrict__ f_node,
                                 int s) {
    int n = blockIdx.x * blockDim.x + threadIdx.x;
    if (n >= NVV) return;
    int b = blockIdx.y;

    size_t cur = (size_t)(b * TRAJ + s) * SLICE + 3 * (size_t)n;
    size_t nxt = cur + SLICE;

    float* fp = f_node + (size_t)b * SLICE + 3 * (size_t)n;
    float fx = fp[0], fy = fp[1], fz = fp[2];
    fp[0] = 0.0f; fp[1] = 0.0f; fp[2] = 0.0f;          // clean for next substep

    float vx = (traj_vel[cur + 0] + DT_C * fx) * DAMP_C;            // MASS == 1
    float vy = (traj_vel[cur + 1] + DT_C * (fy + GRAV_Y)) * DAMP_C;
    float vz = (traj_vel[cur + 2] + DT_C * fz) * DAMP_C;
    float px = traj_pos[cur + 0] + DT_C * vx;
    float py = traj_pos[cur + 1] + DT_C * vy;
    float pz = traj_pos[cur + 2] + DT_C * vz;

    traj_vel[nxt + 0] = vx; traj_vel[nxt + 1] = vy; traj_vel[nxt + 2] = vz;
    traj_pos[nxt + 0] = px; traj_pos[nxt + 1] = py; traj_pos[nxt + 2] = pz;
}

// rest_eff[b,e] = rest_len[e] * (1 + 0.1 * tanh(action[b,e]))  (loop-invariant precompute)
__global__ void rest_eff_kernel(const float* __restrict__ action,
                                const float* __restrict__ rest_len,
                                float* __restrict__ rest_eff) {
    int e = blockIdx.x * blockDim.x + threadIdx.x;
    if (e >= NEE) return;
    int b = blockIdx.y;
    size_t idx = (size_t)b * NEE + e;
    rest_eff[idx] = rest_len[e] * (1.0f + ACT_SCALE * tanhf(action[idx]));
}

// float4 copy of initial state into trajectory slice 0 + zero the force
// accumulator (workspace is poisoned to 0xAA; this keeps every launch deterministic).
__global__ void init_state_kernel(const float4* __restrict__ in_pos,
                                  const float4* __restrict__ in_vel,
                                  float4* __restrict__ out_pos,
                                  float4* __restrict__ out_vel,
                                  float4* __restrict__ f_node) {
    int r = blockIdx.x * blockDim.x + threadIdx.x;
    if (r >= SLICE4) return;
    int b = blockIdx.y;
    size_t src = (size_t)b * SLICE4 + r;
    size_t dst = (size_t)b * TRAJ * SLICE4 + r;    // slice s = 0
    out_pos[dst] = in_pos[src];
    out_vel[dst] = in_vel[src];
    f_node[src]  = make_float4(0.0f, 0.0f, 0.0f, 0.0f);
}

extern "C" void kernel_launch(void* const* d_in, const int* in_sizes, int n_in,
                              void* d_out, int out_size, void* d_ws, size_t ws_size,
                              hipStream_t stream) {
    (void)in_sizes; (void)n_in; (void)out_size; (void)ws_size;

    const float* action   = (const float*)d_in[0];   // [B, NE]
    const float* in_pos   = (const float*)d_in[1];   // [B, NV, 3]
    const float* in_vel   = (const float*)d_in[2];   // [B, NV, 3]
    const float* rest_len = (const float*)d_in[3];   // [NE]
    const int2*  edges    = (const int2*)d_in[4];    // [NE, 2]

    float* traj_pos = (float*)d_out;                             // [B, 11, NV, 3]
    float* traj_vel = traj_pos + (size_t)BB * TRAJ * SLICE;      // [B, 11, NV, 3]

    // Workspace: rest_eff (B*NE f32 = 6.4MB) | f_node (B*NV*3 f32 = 4.8MB)
    float* rest_eff = (float*)d_ws;
    float* f_node   = rest_eff + (size_t)BB * NEE;

    const int TPB = 256;  // 8 wave32 waves per block
    dim3 blk(TPB);
    dim3 gE((NEE + TPB - 1) / TPB, BB);      // per-edge kernels
    dim3 gN((NVV + TPB - 1) / TPB, BB);      // per-node kernel
    dim3 gC((SLICE4 + TPB - 1) / TPB, BB);   // float4 init copy

    rest_eff_kernel<<<gE, blk, 0, stream>>>(action, rest_len, rest_eff);
    init_state_kernel<<<gC, blk, 0, stream>>>((const float4*)in_pos, (const float4*)in_vel,
                                              (float4*)traj_pos, (float4*)traj_vel,
                                              (float4*)f_node);

    for (int s = 0; s < NSUB; ++s) {
        force_kernel<<<gE, blk, 0, stream>>>(traj_pos, edges, rest_eff, f_node, s);
        integrate_kernel<<<gN, blk, 0, stream>>>(traj_pos, traj_vel, f_node, s);
    }
}